// QConv1d_11562051961284
// MI455X (gfx1250) — compile-verified
//
#include <hip/hip_runtime.h>
#include <cstdint>

// Problem constants (from reference): B=2, CIN=16, COUT=16, T=2048, FL=9, stride=1
#define FLT   9
#define NB    2
#define CIN   16
#define COUT  16
#define T_IN  2048
#define T_OUT 2040
#define TT    8               // time tile per block (255 tiles exactly)
#define XT    (TT + FLT - 1)  // 16 staged timesteps per cin (power of two)
#define NW    (COUT * CIN * FLT) // 2304 weights per tensor
#define NSPL  4               // cin-split factor (4 cins per thread)
#define NTHR  512             // 4 cin-groups x 16 couts x 8 t = 16 wave32s

// ---- CDNA5 async global->LDS copy (ASYNCcnt path) ----------------------
#if defined(__has_builtin)
#if __has_builtin(__builtin_amdgcn_global_load_async_to_lds_b128)
#define HAVE_ASYNC_BUILTIN 1
#endif
#endif

typedef int   v4i __attribute__((ext_vector_type(4)));
typedef float f4  __attribute__((ext_vector_type(4)));

typedef __attribute__((address_space(1))) v4i* gv4i_p;  // global (device) ptr
typedef __attribute__((address_space(3))) v4i* lv4i_p;  // LDS ptr

__device__ __forceinline__ void copy16_async(const float* g, float* l) {
#ifdef HAVE_ASYNC_BUILTIN
  __builtin_amdgcn_global_load_async_to_lds_b128(
      (gv4i_p)(unsigned long long)(uintptr_t)g,
      (lv4i_p)(unsigned)(uintptr_t)l,
      0, 0);
#else
  asm volatile("global_load_async_to_lds_b128 %0, %1, off"
               :: "v"((unsigned)(uintptr_t)l), "v"(g) : "memory");
#endif
}

__device__ __forceinline__ void wait_async0() {
#if defined(__has_builtin) && __has_builtin(__builtin_amdgcn_s_wait_asynccnt)
  __builtin_amdgcn_s_wait_asynccnt(0);
#else
  asm volatile("s_wait_asynccnt 0" ::: "memory");
#endif
}

__global__ __launch_bounds__(NTHR) void qconv1d_kernel(
    const float* __restrict__ x, const float* __restrict__ Aw,
    const float* __restrict__ Bw, const float* __restrict__ Cw,
    float* __restrict__ out) {
  __shared__ __align__(16) float sx[CIN * XT * 4];  // [cin][dt][wxyz] 4 KB
  __shared__ __align__(16) float sa[NW];            // 9 KB each
  __shared__ __align__(16) float sb[NW];
  __shared__ __align__(16) float sc[NW];
  __shared__ __align__(16) float sred[NTHR * 4];    // 8 KB partial quaternions

  const int tid = threadIdx.x;
  const int t0  = blockIdx.x * TT;
  const int b   = blockIdx.y;

  // Stage x tile: 16 cin rows x 16 timesteps, one async B128 per chunk.
  // Max global t touched: 254*8 + 15 = 2047 -> always in bounds, no clamp.
  if (tid < CIN * XT) {
    const int cin = tid >> 4;           // XT == 16
    const int k   = tid & (XT - 1);
    copy16_async(&x[(((size_t)b * CIN + cin) * T_IN + t0 + k) * 4], &sx[tid * 4]);
  }
  // Stage a, b, c weights (2304 floats each = 576 x B128 chunks).
  for (int ci = tid; ci < NW / 4; ci += NTHR) {
    copy16_async(&Aw[ci * 4], &sa[ci * 4]);
    copy16_async(&Bw[ci * 4], &sb[ci * 4]);
    copy16_async(&Cw[ci * 4], &sc[ci * 4]);
  }
  wait_async0();
  __syncthreads();

  const int tl   = tid & (TT - 1);          // time within tile: 0..7
  const int cout = (tid >> 3) & (COUT - 1); // 0..15
  const int cg   = tid >> 7;                // cin group: 0..3
  const float* __restrict__ As = &sa[cout * CIN * FLT];
  const float* __restrict__ Bs = &sb[cout * CIN * FLT];
  const float* __restrict__ Cs = &sc[cout * CIN * FLT];

  float aw = 0.f, ax = 0.f, ay = 0.f, az = 0.f;

  const int cin0 = cg * (CIN / NSPL);
  for (int cin = cin0; cin < cin0 + CIN / NSPL; ++cin) {
    const f4* xr = (const f4*)&sx[(cin * XT + tl) * 4]; // tap fl at xr[fl]
    const f4 qp = xr[FLT / 2];                          // center quaternion
    const float pw0 = qp.x, px = qp.y, py = qp.z, pz = qp.w;
    const float pn0 = px * px + py * py + pz * pz;
#pragma unroll
    for (int fl = 0; fl < FLT; ++fl) {
      const f4 qv = xr[fl];
      const float qw = qv.x, qx = qv.y, qy = qv.z, qz = qv.w;
      const int wi = cin * FLT + fl;
      const float cs = Cs[wi], bs = Bs[wi], as_ = As[wi];
      const float pw  = pw0 + cs;          // qpc = qp + c*eye
      const float inv = 1.0f / (pw * pw + pn0);
      // inner = hmul(q, conj(p))
      const float iw =  qw * pw + qx * px + qy * py + qz * pz;
      const float ix = -qw * px + qx * pw - qy * pz + qz * py;
      const float iy = -qw * py + qx * pz + qy * pw - qz * px;
      const float iz = -qw * pz - qx * py + qy * px + qz * pw;
      // r = hmul(p, inner) / |p|^2
      const float rw = (pw * iw - px * ix - py * iy - pz * iz) * inv;
      const float rx = (pw * ix + px * iw + py * iz - pz * iy) * inv;
      const float ry = (pw * iy - px * iz + py * iw + pz * ix) * inv;
      const float rz = (pw * iz + px * iy - py * ix + pz * iw) * inv;
      // e = hmul(q, r) + b*r ; acc += a*e
      const float ew = qw * rw - qx * rx - qy * ry - qz * rz + bs * rw;
      const float ex = qw * rx + qx * rw + qy * rz - qz * ry + bs * rx;
      const float ey = qw * ry - qx * rz + qy * rw + qz * rx + bs * ry;
      const float ez = qw * rz + qx * ry - qy * rx + qz * rw + bs * rz;
      aw += as_ * ew; ax += as_ * ex; ay += as_ * ey; az += as_ * ez;
    }
  }

  // 4-way cross-wave reduction of cin-group partials through LDS.
  f4 part; part.x = aw; part.y = ax; part.z = ay; part.w = az;
  *(f4*)&sred[tid * 4] = part;
  __syncthreads();

  if (cg == 0) {  // 255*8 == 2040, no tail: every (cout, t) is valid
    f4 o = part;
    o += *(const f4*)&sred[(tid + 128) * 4];
    o += *(const f4*)&sred[(tid + 256) * 4];
    o += *(const f4*)&sred[(tid + 384) * 4];
    const int t = t0 + tl;
    *(f4*)&out[(((size_t)b * COUT + cout) * T_OUT + t) * 4] = o;
  }
}

extern "C" void kernel_launch(void* const* d_in, const int* in_sizes, int n_in,
                              void* d_out, int out_size, void* d_ws, size_t ws_size,
                              hipStream_t stream) {
  const float* x = (const float*)d_in[0];
  const float* a = (const float*)d_in[1];
  const float* b = (const float*)d_in[2];
  const float* c = (const float*)d_in[3];
  float* out = (float*)d_out;
  (void)in_sizes; (void)n_in; (void)out_size; (void)d_ws; (void)ws_size;
  dim3 grid(T_OUT / TT, NB);   // 255 x 2 blocks, 16 wave32s each
  qconv1d_kernel<<<grid, NTHR, 0, stream>>>(x, a, b, c, out);
}